// GatingAttention_5016521802181
// MI455X (gfx1250) — compile-verified
//
#include <hip/hip_runtime.h>
#include <hip/hip_bf16.h>

// Problem constants (match the reference)
#define BB 16
#define HH 8
#define SS 512
#define FF 512
#define DD 64
#define RANK 12
#define TOPK 51          // max(1, int(0.1 * 512))
#define LN_EPS 1e-5f
#define SCALE_F 0.044194173824159216f   // 1/sqrt(512)

typedef __attribute__((ext_vector_type(2))) float v2f;
typedef __attribute__((ext_vector_type(8))) float v8f;
typedef __attribute__((ext_vector_type(4))) unsigned int u32x4;
typedef __attribute__((ext_vector_type(8))) int i32x8;
typedef __attribute__((ext_vector_type(4))) int i32x4;

#define FPAD 516   // attn tile row stride (F + 4) -> conflict-free A fragments
#define DPAD 80    // vals tile row stride (D + 16) via TDM pad -> conflict-free B fragments

// ---------------- wave32 helpers ----------------
__device__ __forceinline__ float wsumf(float v) {
    #pragma unroll
    for (int m = 16; m >= 1; m >>= 1) v += __shfl_xor(v, m, 32);
    return v;
}
__device__ __forceinline__ float wmaxf(float v) {
    #pragma unroll
    for (int m = 16; m >= 1; m >>= 1) v = fmaxf(v, __shfl_xor(v, m, 32));
    return v;
}
__device__ __forceinline__ int wsumi(int v) {
    #pragma unroll
    for (int m = 16; m >= 1; m >>= 1) v += __shfl_xor(v, m, 32);
    return v;
}

// map float to monotonically-ordered unsigned key
__device__ __forceinline__ unsigned f2ord(float x) {
    unsigned u = __float_as_uint(x);
    return (u & 0x80000000u) ? ~u : (u | 0x80000000u);
}

// kth-largest key of a 512-element row held as 16 regs/lane across a wave32.
// Bitwise bisection: largest threshold t with count(key >= t) >= k  ==  kth key.
__device__ __forceinline__ unsigned kth_key16(const unsigned* keys, int k) {
    unsigned prefix = 0u;
    for (int bit = 31; bit >= 0; --bit) {
        unsigned cand = prefix | (1u << bit);
        int cnt = 0;
        #pragma unroll
        for (int j = 0; j < 16; ++j) cnt += (keys[j] >= cand) ? 1 : 0;
        cnt = wsumi(cnt);
        if (cnt >= k) prefix = cand;
    }
    return prefix;
}

// ---------------- kernel 1: per-(h,s) precompute ----------------
// alpha_probs[h,s,:] = softmax(topk_mask(alpha*scale));  row_base[h,s,:] = gamma + U.V
__global__ __launch_bounds__(256) void k1_rows(const float* __restrict__ alpha,
                                               const float* __restrict__ gamma,
                                               const float* __restrict__ U,
                                               const float* __restrict__ V,
                                               float* __restrict__ aprob,
                                               float* __restrict__ rbase) {
    const int row  = blockIdx.x * 8 + (threadIdx.x >> 5);   // h*S + s
    const int lane = threadIdx.x & 31;
    const int h = row >> 9;

    float x[16]; unsigned keys[16];
    #pragma unroll
    for (int j = 0; j < 16; ++j) {
        const int f = lane + (j << 5);
        x[j] = alpha[(size_t)row * FF + f] * SCALE_F;
        keys[j] = f2ord(x[j]);
    }
    const unsigned kth = kth_key16(keys, TOPK);

    float m = -1e30f;
    #pragma unroll
    for (int j = 0; j < 16; ++j) m = fmaxf(m, x[j]);
    m = wmaxf(m);

    float p[16]; float sum = 0.f;
    #pragma unroll
    for (int j = 0; j < 16; ++j) {
        p[j] = (keys[j] >= kth) ? __expf(x[j] - m) : 0.f;
        sum += p[j];
    }
    sum = wsumf(sum);
    const float inv = 1.f / sum;
    #pragma unroll
    for (int j = 0; j < 16; ++j)
        aprob[(size_t)row * FF + lane + (j << 5)] = p[j] * inv;

    // rank-12 bilinear + gamma
    float u[RANK];
    #pragma unroll
    for (int r = 0; r < RANK; ++r) u[r] = U[(size_t)row * RANK + r];
    const float g = gamma[row];
    #pragma unroll
    for (int j = 0; j < 16; ++j) {
        const int f = lane + (j << 5);
        float acc = g;
        #pragma unroll
        for (int r = 0; r < RANK; ++r)
            acc += u[r] * V[((size_t)h * RANK + r) * FF + f];
        rbase[(size_t)row * FF + f] = acc;
    }
}

// ---------------- kernel 2: per-(b,h) score (energy -> rms -> gain -> layernorm) ----
__device__ __forceinline__ float blk_sum(float v, float* sred, int tid) {
    v = wsumf(v);
    __syncthreads();
    if ((tid & 31) == 0) sred[tid >> 5] = v;
    __syncthreads();
    float t = 0.f;
    #pragma unroll
    for (int i = 0; i < 8; ++i) t += sred[i];
    return t;
}

__global__ __launch_bounds__(256) void k2_score(const float* __restrict__ values,
                                                const float* __restrict__ temp,
                                                const float* __restrict__ lnw,
                                                const float* __restrict__ lnb,
                                                float* __restrict__ score) {
    const int bh = blockIdx.x;
    const int b = bh >> 3, h = bh & 7;
    const int tid = threadIdx.x;
    __shared__ float e[FF];
    __shared__ float sred[8];

    for (int f = tid; f < FF; f += 256) {
        const float* vp = values + (((size_t)b * FF + f) * HH + h) * DD;
        float s = 0.f;
        #pragma unroll
        for (int d = 0; d < DD; d += 4) {
            const float4 v = *(const float4*)(vp + d);
            s += v.x * v.x + v.y * v.y + v.z * v.z + v.w * v.w;
        }
        e[f] = s * (1.f / DD);
    }
    __syncthreads();

    const float tot = blk_sum(e[tid] + e[tid + 256], sred, tid);
    const float rms = fmaxf(sqrtf(tot * (1.f / FF)), 1e-6f);
    const float t = temp[h];
    const float gain = (t > 20.f) ? t : log1pf(__expf(t));
    const float g = gain / rms;

    const float s0 = e[tid] * g, s1 = e[tid + 256] * g;
    const float mu = blk_sum(s0 + s1, sred, tid) * (1.f / FF);
    const float d0 = s0 - mu, d1 = s1 - mu;
    const float var = blk_sum(d0 * d0 + d1 * d1, sred, tid) * (1.f / FF);
    const float invs = rsqrtf(var + LN_EPS);

    score[(size_t)bh * FF + tid]       = d0 * invs * lnw[tid] + lnb[tid];
    score[(size_t)bh * FF + tid + 256] = d1 * invs * lnw[tid + 256] + lnb[tid + 256];
}

// ---------------- kernel 3: main fused attention (one block per (b,h)) ----------------
// LDS layout (dynamic, base offset 0):
//   valsPad [512][DPAD]  <- TDM-loaded values[b,:,h,:], 16-DWORD pad per 64-DWORD row
//   attn    [32][FPAD]   <- logits -> attn_mix tile
//   scl     [512]        <- score row
__global__ __launch_bounds__(256) void k3_main(const float* __restrict__ values,
                                               const float* __restrict__ aprob,
                                               const float* __restrict__ rbase,
                                               const float* __restrict__ score,
                                               float* __restrict__ out) {
    extern __shared__ float smem[];
    float* valsPad = smem;                     // FF * DPAD floats, MUST sit at LDS offset 0
    float* attn    = valsPad + FF * DPAD;      // 32 * FPAD
    float* scl     = attn + 32 * FPAD;         // 512

    const int bh = blockIdx.x;
    const int b = bh >> 3, h = bh & 7;
    const int tid = threadIdx.x, lane = tid & 31, wave = tid >> 5;

    // ---- Tensor Data Mover: DMA values[b,:,h,:] (512 rows x 64 f32, row stride 512 f32)
    // into LDS with 16-DWORD padding after every 64-DWORD row -> [512][80] layout.
    if (wave == 0) {
        const unsigned long long gaddr =
            (unsigned long long)(const void*)(values + (((size_t)b * FF) * HH + h) * DD);
        // D# group0: count=1 | lds_addr=0 | global_addr | type=2
        u32x4 g0 = {1u,
                    0u,
                    (unsigned)(gaddr & 0xffffffffull),
                    (unsigned)((gaddr >> 32) & 0x1ffffffull) | 0x80000000u};
        // D# group1: data_size=4B(2) | pad_enable | pad_interval=5(64 DW) | pad_amount=15(16 DW)
        //   tensor_dim0=64 (bits79:48), tensor_dim1=512 (bits111:80), tile_dim0=64 (bits127:112)
        //   tile_dim1=512 (bits143:128), tensor_dim0_stride=512 (bits207:160)
        i32x8 g1 = {(int)((2u << 16) | (1u << 20) | (5u << 22) | (15u << 25)),
                    (int)(64u << 16),
                    (int)(512u << 16),
                    (int)(64u << 16),
                    512,
                    512,
                    0, 0};
        i32x4 g2 = {0, 0, 0, 0};
        i32x4 g3 = {0, 0, 0, 0};
        i32x8 g4 = {0, 0, 0, 0, 0, 0, 0, 0};
        __builtin_amdgcn_tensor_load_to_lds(g0, g1, g2, g3, g4, 0);
    }

    for (int f = tid; f < FF; f += 256) scl[f] = score[(size_t)bh * FF + f];

    if (wave == 0) __builtin_amdgcn_s_wait_tensorcnt(0);
    __syncthreads();   // releases waves 1..7 once TDM data landed

    const int sub = wave >> 2;   // which 16-row s-subtile this wave computes
    const int dt  = wave & 3;    // which 16-col d-tile

    for (int s0 = 0; s0 < SS; s0 += 32) {
        // prefetch next s-tile's row_base slab (256 B per thread covers 64 KB)
        if (s0 + 32 < SS)
            __builtin_prefetch((const void*)(rbase + ((size_t)(h * SS + s0 + 32)) * FF + tid * 64), 0, 0);

        // step 1: build 32 logit rows
        for (int i = tid; i < 32 * FF; i += 256) {
            const int r = i >> 9, f = i & 511;
            attn[r * FPAD + f] = scl[f] + rbase[((size_t)(h * SS + s0 + r)) * FF + f];
        }
        __syncthreads();

        // step 2: per-row top-51 mask + softmax + add alpha softmax (4 rows per wave)
        for (int rr = 0; rr < 4; ++rr) {
            const int r = wave * 4 + rr;
            float* rowp = attn + r * FPAD;
            float x[16]; unsigned keys[16];
            #pragma unroll
            for (int j = 0; j < 16; ++j) {
                x[j] = rowp[lane + (j << 5)];
                keys[j] = f2ord(x[j]);
            }
            const unsigned kth = kth_key16(keys, TOPK);
            float m = -1e30f;
            #pragma unroll
            for (int j = 0; j < 16; ++j) m = fmaxf(m, x[j]);
            m = wmaxf(m);
            float p[16]; float sum = 0.f;
            #pragma unroll
            for (int j = 0; j < 16; ++j) {
                p[j] = (keys[j] >= kth) ? __expf(x[j] - m) : 0.f;
                sum += p[j];
            }
            sum = wsumf(sum);
            const float inv = 1.f / sum;
            const float* ap = aprob + ((size_t)(h * SS + s0 + r)) * FF;
            #pragma unroll
            for (int j = 0; j < 16; ++j)
                rowp[lane + (j << 5)] = p[j] * inv + ap[lane + (j << 5)];
        }
        __syncthreads();

        // step 3: 16x16 output tile via V_WMMA_F32_16X16X4_F32, K = 512
        // A frag: lane holds M=lane&15, K = k0 + 2*(lane>>4) .. +1 (contiguous float2)
        // B frag: lane holds N=lane&15, same K pair; valsPad rows DPAD apart
        //   (k vs k+2 rows differ by 160 DW = 32 banks -> conflict-free halves)
        const int hi2 = (lane >> 4) << 1;
        const float* arow  = attn + (sub * 16 + (lane & 15)) * FPAD + hi2;
        const float* bbase = valsPad + hi2 * DPAD + dt * 16 + (lane & 15);
        v8f c = {0.f, 0.f, 0.f, 0.f, 0.f, 0.f, 0.f, 0.f};
        #pragma unroll 8
        for (int k0 = 0; k0 < FF; k0 += 4) {
            const v2f a = *(const v2f*)(arow + k0);
            v2f bv;
            bv.x = bbase[k0 * DPAD];
            bv.y = bbase[k0 * DPAD + DPAD];
            c = __builtin_amdgcn_wmma_f32_16x16x4_f32(false, a, false, bv,
                                                      (short)0, c, false, false);
        }

        // C/D layout: VGPR j -> M = j + 8*(lane>=16), N = lane&15
        const int hi = lane >> 4, n = lane & 15;
        #pragma unroll
        for (int j = 0; j < 8; ++j) {
            const int s = s0 + sub * 16 + j + 8 * hi;
            out[(((size_t)b * SS + s) * HH + h) * DD + dt * 16 + n] = c[j];
        }
        __syncthreads();
    }
}

// ---------------- launcher ----------------
extern "C" void kernel_launch(void* const* d_in, const int* in_sizes, int n_in,
                              void* d_out, int out_size, void* d_ws, size_t ws_size,
                              hipStream_t stream) {
    const float* values = (const float*)d_in[0];
    const float* alpha  = (const float*)d_in[1];
    const float* temp   = (const float*)d_in[2];
    const float* gamma  = (const float*)d_in[3];
    const float* U      = (const float*)d_in[4];
    const float* V      = (const float*)d_in[5];
    const float* lnw    = (const float*)d_in[6];
    const float* lnb    = (const float*)d_in[7];
    float* out = (float*)d_out;

    float* ws    = (float*)d_ws;
    float* aprob = ws;                               // H*S*F
    float* rbase = aprob + (size_t)HH * SS * FF;     // H*S*F
    float* score = rbase + (size_t)HH * SS * FF;     // B*H*F

    const size_t smem3 = (size_t)(FF * DPAD + 32 * FPAD + FF) * sizeof(float);
    (void)hipFuncSetAttribute((const void*)k3_main,
                              hipFuncAttributeMaxDynamicSharedMemorySize, (int)smem3);

    k1_rows<<<(HH * SS) / 8, 256, 0, stream>>>(alpha, gamma, U, V, aprob, rbase);
    k2_score<<<BB * HH, 256, 0, stream>>>(values, temp, lnw, lnb, score);
    k3_main<<<BB * HH, 256, smem3, stream>>>(values, aprob, rbase, score, out);
}